// SmallSMBlock_44753559224601
// MI455X (gfx1250) — compile-verified
//
#include <hip/hip_runtime.h>

typedef __attribute__((ext_vector_type(2))) float v2f;
typedef __attribute__((ext_vector_type(8))) float v8f;

#define IMG_N   1024
#define TILE_W  128
#define IMG_STR 132     // padded LDS row stride in floats (132 % 64 banks = 4)

// Compile-time A-operand LDS offset for reduction index k = c*9 + u*3 + v.
// k == 27 (pad) points at the all-zero LDS row.
__device__ constexpr int a_off(int k) {
    return (k < 27)
        ? (((k / 9) * 3 + ((k % 9) / 3)) * IMG_STR + (k % 3))
        : (9 * IMG_STR);
}

// Fused per-pixel adaptive stencil:
//   Phase 1 (WMMA): K[9, row, col0:col0+128] = im2col(image) x weights^T + bias
//     per wave: M=16 pixels, N=16 (9 taps + pad), K=28 (27 + pad)
//     -> 7 chained v_wmma_f32_16x16x4_f32; A from a pre-padded LDS image tile,
//     B from pre-padded LDS weights -> all operand fetches are unconditional.
//   Phase 2 (VALU): y[b,row,j] = sum_n K[n,row,j] * x[b,row-1+n/3, j-1+n%3]
//     edge handling folded into K coefficients (branchless); x re-reads hit L2.
__global__ __launch_bounds__(256) void fused_stencil_kernel(
    const float* __restrict__ image,
    const float* __restrict__ x,
    const float* __restrict__ weights,
    const float* __restrict__ bias,
    float* __restrict__ y)
{
    __shared__ float lds_wp[28 * 16];        // B: [k][n], zero-padded
    __shared__ float lds_b[16];              // bias, zero-padded
    __shared__ float lds_img[10 * IMG_STR];  // rows 0..8 = (c*3+u); row 9 = zeros
    __shared__ float lds_K[9][IMG_STR];      // K taps for this block's 128 pixels

    const int row  = blockIdx.y;             // scalar across block
    const int col0 = blockIdx.x * TILE_W;
    const int t    = threadIdx.x;

    // Warm L2 for this block's x stream while phase 1 runs (global_prefetch_b8).
    {
        const int pj = col0 + (t & 127);
        const int pb = (t >> 7) * 8;
        #pragma unroll
        for (int bb = 0; bb < 8; ++bb)
            __builtin_prefetch(x + (size_t)(pb + bb) * IMG_N * IMG_N
                                 + (size_t)row * IMG_N + pj, 0, 1);
    }

    // ---- stage zero-padded weights: lds_wp[k*16 + n] = weights[n*27 + k] ----
    for (int i = t; i < 28 * 16; i += 256) {
        const int n = i & 15;
        const int k = i >> 4;
        lds_wp[i] = (n < 9 && k < 27) ? weights[n * 27 + k] : 0.0f;
    }
    if (t < 16) lds_b[t] = (t < 9) ? bias[t] : 0.0f;

    // ---- stage image halo tile: 3ch x 3rows x 130cols (+ zero row 9) ----
    for (int i = t; i < 10 * IMG_STR; i += 256) {
        const int rr   = i / IMG_STR;
        const int slot = i - rr * IMG_STR;
        const int c    = rr / 3;
        const int u    = rr - 3 * c;
        const int ii   = row - 1 + u;
        const int jj   = col0 - 1 + slot;
        float v = 0.0f;
        if (rr < 9 && slot < 130 && ii >= 0 && ii < IMG_N && jj >= 0 && jj < IMG_N)
            v = image[((size_t)c * IMG_N + ii) * IMG_N + jj];
        lds_img[i] = v;
    }
    __syncthreads();

    // ---------------- Phase 1: K via WMMA f32 16x16x4 ----------------
    const int wave  = t >> 5;
    const int lane  = t & 31;
    const int mlane = lane & 15;            // A: M (pixel); B/C/D: N (tap)
    const int hi    = lane >> 4;            // upper half-wave holds K+2
    const int pbase = wave * 16 + mlane;    // A pixel slot (a_off supplies +v)
    const int bbase = hi * 32 + mlane;      // B elem base: (2*hi)*16 + n

    v8f acc;
    {
        const float bv = lds_b[mlane];      // C[m][n] = bias[n] (0 for n>=9)
        #pragma unroll
        for (int r = 0; r < 8; ++r) acc[r] = bv;
    }

    #pragma unroll
    for (int s = 0; s < 7; ++s) {
        v2f a, b;
        #pragma unroll
        for (int sub = 0; sub < 2; ++sub) {
            // koff: compile-time constants selected by hi (one v_cndmask)
            const int koff = hi ? a_off(4 * s + 2 + sub) : a_off(4 * s + sub);
            a[sub] = lds_img[koff + pbase];
            b[sub] = lds_wp[bbase + (4 * s + sub) * 16];
        }
        // D = A(16x4) x B(4x16) + C
        acc = __builtin_amdgcn_wmma_f32_16x16x4_f32(
            false, a, false, b, (short)0, acc, false, false);
    }

    // D layout: vgpr r, lane l -> M = r + 8*hi (pixel-in-wave), N = mlane (tap)
    if (mlane < 9) {
        const int base = wave * 16 + 8 * hi;
        #pragma unroll
        for (int r = 0; r < 8; ++r)
            lds_K[mlane][base + r] = acc[r];
    }
    __syncthreads();

    // -------- Phase 2: apply per-pixel stencil over 16 batches (branchless) --------
    const int px = t & 127;
    const int j  = col0 + px;
    const int b0 = (t >> 7) * 8;            // threads 0..127 -> b 0..7; 128..255 -> 8..15

    const bool okl = (j > 0);
    const bool okr = (j < IMG_N - 1);
    const int  jl  = okl ? (j - 1) : 0;     // clamped (safe) addresses
    const int  jr  = okr ? (j + 1) : (IMG_N - 1);

    // Fold column-edge zeroing into the coefficients: 6 cndmasks total.
    float Km[9];
    #pragma unroll
    for (int u = 0; u < 3; ++u) {
        Km[u * 3 + 0] = okl ? lds_K[u * 3 + 0][px] : 0.0f;
        Km[u * 3 + 1] =       lds_K[u * 3 + 1][px];
        Km[u * 3 + 2] = okr ? lds_K[u * 3 + 2][px] : 0.0f;
    }

    #pragma unroll
    for (int bb = 0; bb < 8; ++bb) {
        const int b = b0 + bb;
        const float* __restrict__ xb = x + (size_t)b * IMG_N * IMG_N;
        float accy = 0.0f;
        #pragma unroll
        for (int u = 0; u < 3; ++u) {
            const int ii = row - 1 + u;
            if (ii >= 0 && ii < IMG_N) {    // scalar branch (row uniform)
                const float* __restrict__ xr = xb + (size_t)ii * IMG_N;
                accy = fmaf(Km[u * 3 + 0], xr[jl], accy);
                accy = fmaf(Km[u * 3 + 1], xr[j],  accy);
                accy = fmaf(Km[u * 3 + 2], xr[jr], accy);
            }
        }
        y[(size_t)b * IMG_N * IMG_N + (size_t)row * IMG_N + j] = accy;
    }
}

extern "C" void kernel_launch(void* const* d_in, const int* in_sizes, int n_in,
                              void* d_out, int out_size, void* d_ws, size_t ws_size,
                              hipStream_t stream) {
    const float* image   = (const float*)d_in[0];  // [3,1024,1024]
    const float* x       = (const float*)d_in[1];  // [16,1,1024,1024]
    const float* weights = (const float*)d_in[2];  // [9,3,3,3]
    const float* bias    = (const float*)d_in[3];  // [9]
    float* y = (float*)d_out;                      // [16,1,1024,1024]

    dim3 grid(IMG_N / TILE_W, IMG_N, 1);
    fused_stencil_kernel<<<grid, 256, 0, stream>>>(image, x, weights, bias, y);
}